// EnhancedBilinearInteraction_49134425866945
// MI455X (gfx1250) — compile-verified
//
#include <hip/hip_runtime.h>

// ---------------------------------------------------------------------------
// EnhancedBilinearInteraction for MI455X (gfx1250)
//
// out[b,m] = sum_l tanh(bn_x(x[b,l,m])) * tanh(bn_y(y[b,l,m]))
// bn stats per feature m over (B,L); biased variance; eps=1e-5.
//
// Memory-bound: 1 GiB inputs read twice (stats pass + fused pass) = 2 GiB
// -> ~88us floor at 23.3 TB/s. Everything else is built to saturate HBM:
//  - pass1: float4 nontemporal streaming reduce, deterministic 2-stage
//  - pass2: async global->LDS (B128, double buffered, ASYNCcnt) staging,
//           tanh on VALU, L-reduction folded into V_WMMA_F32_16X16X4_F32
//           (diag of A^T*B trick, full fp32 numerics)
// ---------------------------------------------------------------------------

typedef float f4  __attribute__((ext_vector_type(4)));
typedef float v2f __attribute__((ext_vector_type(2)));
typedef float v8f __attribute__((ext_vector_type(8)));
typedef int   i4  __attribute__((vector_size(16)));   // 128-bit payload type for async LDS builtin

#define B_    64
#define L_    8192
#define M_    256
#define EPS_  1e-5f

// pass 1: stats partials
#define NB1   512                   // blocks
#define ROWS1 ((B_ * L_) / NB1)     // 1024 rows per block

// pass 2: fused main
#define NCH   32                    // L chunks per b  -> 64*32 = 2048 blocks
#define LC    (L_ / NCH)            // 256 rows per block
#define TR    8                     // rows per LDS tile
#define TSTR  264                   // padded row stride in floats (16B-aligned rows, bank-shifted)
#define NTILE (LC / TR)             // 32 tiles per chunk

#if __has_builtin(__builtin_amdgcn_wmma_f32_16x16x4_f32)
#define HAVE_WMMA 1
#else
#define HAVE_WMMA 0
#endif

#if __has_builtin(__builtin_amdgcn_global_load_async_to_lds_b128)
#define HAVE_ASYNC 1
#else
#define HAVE_ASYNC 0
#endif

__device__ __forceinline__ float fast_tanh(float v) {
#if __has_builtin(__builtin_amdgcn_tanhf)
  return __builtin_amdgcn_tanhf(v);   // gfx1250 V_TANH_F32 (trans pipe)
#else
  return tanhf(v);
#endif
}

#if HAVE_ASYNC
// Signature (from clang diagnostic): param0 = i4 addrspace(1)* (prints as
// "int __vector(4) __device__ *"), so: (global src, lds dst, imm off, imm cpol).
typedef __attribute__((address_space(1))) i4 gi4;
typedef __attribute__((address_space(3))) i4 li4;

__device__ __forceinline__ void async_b128(const float* g, float* l) {
  __builtin_amdgcn_global_load_async_to_lds_b128((gi4*)g, (li4*)l, 0, 0);
}
#endif

template <int N>
__device__ __forceinline__ void wait_async_le() {
#if __has_builtin(__builtin_amdgcn_s_wait_asynccnt)
  __builtin_amdgcn_s_wait_asynccnt(N);    // ICE required
#elif HAVE_ASYNC
  asm volatile("s_wait_asynccnt %0" :: "n"(N) : "memory");
#endif
}

// ---------------------------------------------------------------------------
// Pass 1: per-block partial sums of x, x^2, y, y^2 per feature m.
// Deterministic (fixed-order, no atomics). 512 blocks * 4KB partials -> ws.
// ---------------------------------------------------------------------------
__global__ __launch_bounds__(256) void stats_partial(
    const float* __restrict__ x, const float* __restrict__ y,
    float* __restrict__ wsA) {
  const int tid = threadIdx.x;
  const int r   = tid >> 6;        // 0..3 : row group
  const int mq  = tid & 63;        // float4 column (m = mq*4 .. +3)
  const size_t row0 = (size_t)blockIdx.x * ROWS1;

  f4 sx  = {0.f, 0.f, 0.f, 0.f}, sxx = {0.f, 0.f, 0.f, 0.f};
  f4 sy  = {0.f, 0.f, 0.f, 0.f}, syy = {0.f, 0.f, 0.f, 0.f};

  for (int it = 0; it < ROWS1 / 4; ++it) {
    size_t row = row0 + (size_t)it * 4 + r;
    size_t off = row * M_ + (size_t)mq * 4;
    f4 vx = __builtin_nontemporal_load((const f4*)(x + off));
    f4 vy = __builtin_nontemporal_load((const f4*)(y + off));
    sx += vx; sxx += vx * vx;
    sy += vy; syy += vy * vy;
  }

  __shared__ f4 red[256];
  float* o = wsA + (size_t)blockIdx.x * 4 * M_;
  f4 acc[4] = {sx, sxx, sy, syy};
#pragma unroll
  for (int s = 0; s < 4; ++s) {
    red[tid] = acc[s];
    __syncthreads();
    if (r == 0) {
      f4 t = red[mq] + red[64 + mq] + red[128 + mq] + red[192 + mq];
      *(f4*)&o[s * M_ + mq * 4] = t;
    }
    __syncthreads();
  }
}

// ---------------------------------------------------------------------------
// Pass 1b: finalize per-m scale/shift for x and y:
//   norm(t) = t*scale + shift, scale = gamma*rsqrt(var+eps), shift = beta - mean*scale
// ---------------------------------------------------------------------------
__global__ __launch_bounds__(256) void stats_final(
    const float* __restrict__ wsA,
    const float* __restrict__ gamma, const float* __restrict__ beta,
    float* __restrict__ wsB) {
  const int m = threadIdx.x;
  float sx = 0.f, sxx = 0.f, sy = 0.f, syy = 0.f;
  for (int blk = 0; blk < NB1; ++blk) {
    const float* p = wsA + (size_t)blk * 4 * M_;
    sx  += p[0 * M_ + m];
    sxx += p[1 * M_ + m];
    sy  += p[2 * M_ + m];
    syy += p[3 * M_ + m];
  }
  const float invN = 1.0f / (float)((size_t)B_ * L_);
  float mx = sx * invN, vx = fmaxf(sxx * invN - mx * mx, 0.f);
  float my = sy * invN, vy = fmaxf(syy * invN - my * my, 0.f);
  float g = gamma[m], be = beta[m];
  float scx = g * rsqrtf(vx + EPS_);
  float scy = g * rsqrtf(vy + EPS_);
  wsB[0 * M_ + m] = scx;
  wsB[1 * M_ + m] = fmaf(-mx, scx, be);
  wsB[2 * M_ + m] = scy;
  wsB[3 * M_ + m] = fmaf(-my, scy, be);
}

#if HAVE_WMMA
__device__ __forceinline__ v8f wmma_acc(v2f a, v2f b, v8f c) {
  // (neg_a, A, neg_b, B, c_mod, C, reuse_a, reuse_b)
  return __builtin_amdgcn_wmma_f32_16x16x4_f32(false, a, false, b,
                                               (short)0, c, false, false);
}
// D/C 16x16 f32 layout: vgpr r, lanes 0-15 -> (M=r, N=lane); lanes 16-31 -> (M=r+8, N=lane-16)
// diag (i,i): i<8  -> vgpr i,   lane i          (lanes 0..7)
//             i>=8 -> vgpr i-8, lane 16+i       (lanes 24..31)
__device__ __forceinline__ void store_diag(v8f c, int m0, int lane,
                                           float* __restrict__ outp) {
  int r = -1, i = 0;
  if (lane < 8)        { r = lane;      i = lane; }
  else if (lane >= 24) { r = lane - 24; i = lane - 16; }
  if (r >= 0) {
    float d;
    switch (r) {
      case 0: d = c[0]; break; case 1: d = c[1]; break;
      case 2: d = c[2]; break; case 3: d = c[3]; break;
      case 4: d = c[4]; break; case 5: d = c[5]; break;
      case 6: d = c[6]; break; default: d = c[7]; break;
    }
    outp[m0 + i] = d;
  }
}
#endif

// ---------------------------------------------------------------------------
// Pass 2: fused normalize+tanh+product+L-reduce.
// Block = (chunk ch, batch b), 256 threads = 8 waves. Waves own 2 m-tiles of 16.
// Tiles of 8 rows x 256 m staged global->LDS with async B128 (double buffered).
// WMMA: A[i,k]=tx(l0+k, m0+i), B[k,j]=ty(l0+k, m0+j); diag(C) accumulates
// sum_l tx*ty per m. A and B share the identical lane layout (lane=m-in-tile,
// half/vgpr = k), so each lane feeds the 4 tanh values it just computed.
// ---------------------------------------------------------------------------
__global__ __launch_bounds__(256) void fused_main(
    const float* __restrict__ x, const float* __restrict__ y,
    const float* __restrict__ wsB, float* __restrict__ wsC) {
  __shared__ __align__(16) float lx[2][TR * TSTR];
  __shared__ __align__(16) float ly[2][TR * TSTR];

  const int ch = blockIdx.x, b = blockIdx.y;
  const int tid = threadIdx.x, lane = tid & 31, wave = tid >> 5;
  const size_t base = ((size_t)b * L_ + (size_t)ch * LC) * M_;
  const float* gx = x + base;
  const float* gy = y + base;

  const int ml  = lane & 15;
  const int k0  = (lane >> 4) << 1;      // 0 or 2
  const int m0a = wave * 32;
  const int m0b = wave * 32 + 16;
  const float sxa = wsB[0 * M_ + m0a + ml], hxa = wsB[1 * M_ + m0a + ml];
  const float sya = wsB[2 * M_ + m0a + ml], hya = wsB[3 * M_ + m0a + ml];
  const float sxb = wsB[0 * M_ + m0b + ml], hxb = wsB[1 * M_ + m0b + ml];
  const float syb = wsB[2 * M_ + m0b + ml], hyb = wsB[3 * M_ + m0b + ml];

#if HAVE_WMMA
  v8f ca = {}; v8f cb = {};
#else
  float ca = 0.f, cb = 0.f;
#endif

  auto load_tile = [&](int buf, int t) {
    const float* bx = gx + (size_t)t * TR * M_;
    const float* by = gy + (size_t)t * TR * M_;
    // 8 rows * 64 float4 = 512 float4 per matrix; 2 per thread per matrix
#pragma unroll
    for (int j = 0; j < 2; ++j) {
      int f  = tid + j * 256;
      int r  = f >> 6, c4 = f & 63;
      size_t go = (size_t)r * M_ + (size_t)c4 * 4;
      int    lo = r * TSTR + c4 * 4;
#if HAVE_ASYNC
      async_b128(bx + go, &lx[buf][lo]);
      async_b128(by + go, &ly[buf][lo]);
#else
      f4 vx = __builtin_nontemporal_load((const f4*)(bx + go));
      f4 vy = __builtin_nontemporal_load((const f4*)(by + go));
      *(f4*)&lx[buf][lo] = vx;
      *(f4*)&ly[buf][lo] = vy;
#endif
    }
  };

  load_tile(0, 0);
  for (int t = 0; t < NTILE; ++t) {
    const int cur = t & 1;
    const bool more = (t + 1 < NTILE);
    if (more) load_tile(cur ^ 1, t + 1);   // prev-iter trailing barrier guards overwrite
#if HAVE_ASYNC
    if (more) wait_async_le<4>();          // our 4 older asyncs (tile `cur`) done
    else      wait_async_le<0>();
#endif
    __syncthreads();                       // everyone's loads for `cur` visible

#pragma unroll
    for (int kk = 0; kk < TR / 4; ++kk) {
      const int rowb = (kk * 4 + k0) * TSTR;
      {
        const int ia = rowb + m0a + ml;
        float x0 = lx[cur][ia], x1 = lx[cur][ia + TSTR];
        float y0 = ly[cur][ia], y1 = ly[cur][ia + TSTR];
        v2f av = { fast_tanh(fmaf(x0, sxa, hxa)), fast_tanh(fmaf(x1, sxa, hxa)) };
        v2f bv = { fast_tanh(fmaf(y0, sya, hya)), fast_tanh(fmaf(y1, sya, hya)) };
#if HAVE_WMMA
        ca = wmma_acc(av, bv, ca);
#else
        ca = fmaf(av.x, bv.x, ca); ca = fmaf(av.y, bv.y, ca);
#endif
      }
      {
        const int ib = rowb + m0b + ml;
        float x0 = lx[cur][ib], x1 = lx[cur][ib + TSTR];
        float y0 = ly[cur][ib], y1 = ly[cur][ib + TSTR];
        v2f av = { fast_tanh(fmaf(x0, sxb, hxb)), fast_tanh(fmaf(x1, sxb, hxb)) };
        v2f bv = { fast_tanh(fmaf(y0, syb, hyb)), fast_tanh(fmaf(y1, syb, hyb)) };
#if HAVE_WMMA
        cb = wmma_acc(av, bv, cb);
#else
        cb = fmaf(av.x, bv.x, cb); cb = fmaf(av.y, bv.y, cb);
#endif
      }
    }
    __syncthreads();                       // compute(cur) done before it is overwritten
  }

  float* outp = wsC + (size_t)(ch * B_ + b) * M_;
#if HAVE_WMMA
  store_diag(ca, m0a, lane, outp);
  store_diag(cb, m0b, lane, outp);
#else
  float ta = ca + __shfl_xor(ca, 16, 32);  // combine k-halves
  float tb = cb + __shfl_xor(cb, 16, 32);
  if (lane < 16) { outp[m0a + lane] = ta; outp[m0b + lane] = tb; }
#endif
}

// ---------------------------------------------------------------------------
// Pass 2b: deterministic reduce of chunk partials -> out (64 x 256)
// ---------------------------------------------------------------------------
__global__ __launch_bounds__(256) void reduce_out(
    const float* __restrict__ wsC, float* __restrict__ out) {
  const int m = threadIdx.x, b = blockIdx.x;
  float acc = 0.f;
  for (int chh = 0; chh < NCH; ++chh)
    acc += wsC[(size_t)(chh * B_ + b) * M_ + m];
  out[b * M_ + m] = acc;
}

// ---------------------------------------------------------------------------
// Workspace layout (floats):
//   wsA: NB1*4*M_   = 524288  (2 MiB)  stats partials
//   wsB: 4*M_       = 1024             scale/shift
//   wsC: NCH*B_*M_  = 524288  (2 MiB)  chunk partials
// total ~4.2 MB
// ---------------------------------------------------------------------------
extern "C" void kernel_launch(void* const* d_in, const int* in_sizes, int n_in,
                              void* d_out, int out_size, void* d_ws, size_t ws_size,
                              hipStream_t stream) {
  const float* x     = (const float*)d_in[0];
  const float* y     = (const float*)d_in[1];
  const float* gamma = (const float*)d_in[2];
  const float* beta  = (const float*)d_in[3];
  float* ws  = (float*)d_ws;
  float* wsA = ws;
  float* wsB = wsA + (size_t)NB1 * 4 * M_;
  float* wsC = wsB + 4 * M_;
  float* out = (float*)d_out;

  stats_partial<<<NB1, 256, 0, stream>>>(x, y, wsA);
  stats_final<<<1, 256, 0, stream>>>(wsA, gamma, beta, wsB);
  fused_main<<<dim3(NCH, B_), 256, 0, stream>>>(x, y, wsB, wsC);
  reduce_out<<<B_, 256, 0, stream>>>(wsC, out);
}